// PatchMSSSIMLoss_5403068858675
// MI455X (gfx1250) — compile-verified
//
#include <hip/hip_runtime.h>

// ---------------------------------------------------------------------------
// Patch MS-SSIM loss on MI455X (gfx1250, wave32).
// Separable 11-tap Gaussian conv expressed as banded matmul on
// V_WMMA_F32_16X16X4_F32 (fp32 end-to-end). One wave per 32x32 patch.
// ---------------------------------------------------------------------------

typedef __attribute__((ext_vector_type(2))) float v2f;
typedef __attribute__((ext_vector_type(4))) float vf4;
typedef __attribute__((ext_vector_type(8))) float v8f;

#define IMG_W   512
#define NPP     961            // 31*31 patches per (b,c) plane
#define NPATCH  46128          // 16*3*31*31

static __device__ __forceinline__ v8f wmma4(v2f a, v2f b, v8f c) {
  // D = A(16x4) * B(4x16) + C(16x16), fp32. 8-arg VOP3P pattern:
  // (neg_a, A, neg_b, B, c_mod, C, reuse_a, reuse_b)
  return __builtin_amdgcn_wmma_f32_16x16x4_f32(false, a, false, b, (short)0, c,
                                               false, false);
}

static __device__ __forceinline__ float wave_sum(float v) {
#pragma unroll
  for (int o = 16; o; o >>= 1) v += __shfl_xor(v, o, 32);
  return v;
}

static __device__ __forceinline__ void build_g(float* g, int ws) {
  int c = ws / 2;
  float s = 0.f;
  for (int i = 0; i < ws; ++i) {
    float d = (float)(i - c);
    float e = expf(-d * d / 4.5f);   // sigma = 1.5 -> 2*sigma^2 = 4.5
    g[i] = e; s += e;
  }
  float inv = 1.f / s;
  for (int i = 0; i < ws; ++i) g[i] *= inv;
}

// Banded conv matrix element Wh[k][c] = g[k-c], zero outside band / past cmax.
static __device__ __forceinline__ float wband(int k, int c, int cmax, const float* g11) {
  if (c >= cmax) return 0.f;
  int d = k - c;
  return (d >= 0 && d < 11) ? g11[d] : 0.f;
}

// plane 0: A; 1: B; 2: A*A; 3: B*B; 4: A*B
static __device__ __forceinline__ v2f plane_val(int pl, float a1, float a2,
                                                float b1, float b2) {
  v2f r;
  switch (pl) {
    case 0:  r.x = a1;    r.y = a2;    break;
    case 1:  r.x = b1;    r.y = b2;    break;
    case 2:  r.x = a1*a1; r.y = a2*a2; break;
    case 3:  r.x = b1*b1; r.y = b2*b2; break;
    default: r.x = a1*b1; r.y = a2*b2; break;
  }
  return r;
}

static __device__ __forceinline__ void pool2(const float* src, int ss,
                                             float* dst, int ds, int n, int lane) {
  for (int idx = lane; idx < n * n; idx += 32) {
    int r = idx / n, c = idx - r * n;
    dst[r * ds + c] = 0.25f * (src[(2*r)*ss + 2*c]   + src[(2*r)*ss + 2*c + 1] +
                               src[(2*r+1)*ss + 2*c] + src[(2*r+1)*ss + 2*c + 1]);
  }
}

// SSIM for levels where conv 'VALID' output is 1x1 (ws == image size).
static __device__ __forceinline__ void ssim_small(const float* pa, const float* pb,
                                                  int stride, int n, const float* g,
                                                  int lane, float C1, float C2,
                                                  float& ssim_out, float& cs_out) {
  float mu1=0.f, mu2=0.f, m11=0.f, m22=0.f, m12=0.f;
  for (int idx = lane; idx < n*n; idx += 32) {
    int i = idx / n, j = idx - i*n;
    float w = g[i]*g[j];
    float a = pa[i*stride+j], b = pb[i*stride+j];
    mu1 += w*a; mu2 += w*b; m11 += w*a*a; m22 += w*b*b; m12 += w*a*b;
  }
  mu1 = wave_sum(mu1); mu2 = wave_sum(mu2);
  m11 = wave_sum(m11); m22 = wave_sum(m22); m12 = wave_sum(m12);
  float s1 = m11 - mu1*mu1, s2 = m22 - mu2*mu2, s12 = m12 - mu1*mu2;
  float v1 = 2.f*s12 + C2, v2 = s1 + s2 + C2;
  cs_out   = v1 / v2;
  ssim_out = (2.f*mu1*mu2 + C1) * v1 / ((mu1*mu1 + mu2*mu2 + C1) * v2);
}

// ---------------------------------------------------------------------------
// Kernel 0: zero scratch (10 doubles sums + 10 int flags)
// ---------------------------------------------------------------------------
__global__ void k_init(double* sums, int* flags) {
  int t = threadIdx.x;
  if (t < 10) { sums[t] = 0.0; flags[t] = 0; }
}

// ---------------------------------------------------------------------------
// Kernel 1: Lr flags per level from y-patch pyramid.
// flags[l]   = any pooled value > 128
// flags[5+l] = any pooled value < -0.5
// ---------------------------------------------------------------------------
static __device__ __forceinline__ void flag_check(const float* p, int stride, int n,
                                                  int lane, int* hi, int* lo) {
  int h = 0, l = 0;
  for (int idx = lane; idx < n*n; idx += 32) {
    float v = p[(idx / n) * stride + (idx % n)];
    h |= (v > 128.f); l |= (v < -0.5f);
  }
  unsigned long long bh = __ballot(h);
  unsigned long long bl = __ballot(l);
  if (lane == 0) {
    if (bh) atomicOr(hi, 1);
    if (bl) atomicOr(lo, 1);
  }
}

__global__ __launch_bounds__(32)
void k_flags(const float* __restrict__ yimg, int* __restrict__ flags) {
  __shared__ float bufA[32][36];
  __shared__ float pA1[16][18];
  __shared__ float pA2[8][10];
  __shared__ float pA3[4][6];
  __shared__ float pA4[2][4];

  int lane = threadIdx.x;
  int p = blockIdx.x;
  if (p >= NPATCH) return;
  int img = p / NPP, pid = p % NPP;
  int r0 = (pid / 31) * 16, c0 = (pid % 31) * 16;

  const vf4* src = (const vf4*)(yimg + ((size_t)img * IMG_W + r0) * IMG_W + c0 +
                                (size_t)lane * IMG_W);
#pragma unroll
  for (int k = 0; k < 8; ++k) *(vf4*)&bufA[lane][4*k] = src[k];
  __syncthreads();

  flag_check(&bufA[0][0], 36, 32, lane, &flags[0], &flags[5]);
  pool2(&bufA[0][0], 36, &pA1[0][0], 18, 16, lane); __syncthreads();
  flag_check(&pA1[0][0], 18, 16, lane, &flags[1], &flags[6]);
  pool2(&pA1[0][0], 18, &pA2[0][0], 10, 8, lane); __syncthreads();
  flag_check(&pA2[0][0], 10, 8, lane, &flags[2], &flags[7]);
  pool2(&pA2[0][0], 10, &pA3[0][0], 6, 4, lane); __syncthreads();
  flag_check(&pA3[0][0], 6, 4, lane, &flags[3], &flags[8]);
  pool2(&pA3[0][0], 6, &pA4[0][0], 4, 2, lane); __syncthreads();
  flag_check(&pA4[0][0], 4, 2, lane, &flags[4], &flags[9]);
}

// ---------------------------------------------------------------------------
// Kernel 2: main per-patch MS-SSIM pass. One wave per patch.
// sums[0..4] += sum(ssim_map) per level; sums[5..9] += sum(v1/v2) per level.
// ---------------------------------------------------------------------------
__global__ __launch_bounds__(32)
void k_main(const float* __restrict__ yimg, const float* __restrict__ ypred,
            double* __restrict__ sums, const int* __restrict__ flags) {
  __shared__ float bufA[32][36];           // y patch (img1)
  __shared__ float bufB[32][36];           // y_pred patch (img2)
  __shared__ float bufT[32][36];           // H-pass result T = S*Wh
  __shared__ float bufR[5][22][24];        // (S*Wh)^T*Wh per plane (22x22 valid)
  __shared__ float pA1[16][18], pB1[16][18];
  __shared__ float pA2[8][10],  pB2[8][10];
  __shared__ float pA3[4][6],   pB3[4][6];
  __shared__ float pA4[2][4],   pB4[2][4];
  __shared__ float g11s[11], g8s[8], g4s[4], g2s[2];

  int lane = threadIdx.x;
  int p = blockIdx.x;
  if (p >= NPATCH) return;

  if (lane == 0) {
    build_g(g11s, 11); build_g(g8s, 8); build_g(g4s, 4); build_g(g2s, 2);
  }

  // SSIM constants per level from Lr flags.
  float C1v[5], C2v[5];
#pragma unroll
  for (int l = 0; l < 5; ++l) {
    float Lr = (flags[l] ? 255.f : 1.f) + (flags[5 + l] ? 1.f : 0.f);
    float q = Lr * Lr;
    C1v[l] = 1e-4f * q;     // (0.01*Lr)^2
    C2v[l] = 9e-4f * q;     // (0.03*Lr)^2
  }

  // Load both 32x32 patches (lane = patch row).
  int img = p / NPP, pid = p % NPP;
  int r0 = (pid / 31) * 16, c0 = (pid % 31) * 16;
  const vf4* sa = (const vf4*)(yimg  + ((size_t)img * IMG_W + r0) * IMG_W + c0 +
                               (size_t)lane * IMG_W);
  const vf4* sb = (const vf4*)(ypred + ((size_t)img * IMG_W + r0) * IMG_W + c0 +
                               (size_t)lane * IMG_W);
#pragma unroll
  for (int k = 0; k < 8; ++k) {
    *(vf4*)&bufA[lane][4*k] = sa[k];
    *(vf4*)&bufB[lane][4*k] = sb[k];
  }
  __syncthreads();

  int half = lane >> 4;      // K-pair select per ISA A/B 16x4 layout
  int nlo  = lane & 15;      // M row (A) / N col (B, C/D)

  // Constant banded-Gaussian B-operand tiles (per ISA B layout:
  // VGPR v, lanes0-15 -> K=kbase+v, lanes16-31 -> K=kbase+v+2, N=lane&15).
  v2f wb0[8][2];
#pragma unroll
  for (int kc = 0; kc < 8; ++kc)
#pragma unroll
    for (int nt = 0; nt < 2; ++nt) {
      int k0 = kc * 4 + half * 2, c = nt * 16 + nlo;
      v2f t; t.x = wband(k0, c, 22, g11s); t.y = wband(k0 + 1, c, 22, g11s);
      wb0[kc][nt] = t;
    }
  v2f wb1[4];
#pragma unroll
  for (int kc = 0; kc < 4; ++kc) {
    int k0 = kc * 4 + half * 2;
    v2f t; t.x = wband(k0, nlo, 6, g11s); t.y = wband(k0 + 1, nlo, 6, g11s);
    wb1[kc] = t;
  }

  // ----- Level 0: 32x32 -> 22x22 conv via WMMA -----
  for (int pl = 0; pl < 5; ++pl) {
    // H-pass: T = S * Wh  (S built on the fly from bufA/bufB)
#pragma unroll
    for (int mt = 0; mt < 2; ++mt)
#pragma unroll
      for (int nt = 0; nt < 2; ++nt) {
        v8f acc = (v8f)(0.f);
#pragma unroll
        for (int kc = 0; kc < 8; ++kc) {
          int row = mt * 16 + nlo, k0 = kc * 4 + half * 2;
          float a1 = bufA[row][k0], a2 = bufA[row][k0 + 1];
          float b1 = bufB[row][k0], b2 = bufB[row][k0 + 1];
          acc = wmma4(plane_val(pl, a1, a2, b1, b2), wb0[kc][nt], acc);
        }
#pragma unroll
        for (int v = 0; v < 8; ++v)
          bufT[mt * 16 + v + half * 8][nt * 16 + nlo] = acc[v];
      }
    __syncthreads();
    // V-pass: R = T^T * Wh (transpose read from LDS); result stored transposed,
    // which is irrelevant for the subsequent elementwise + mean reduction.
#pragma unroll
    for (int mt = 0; mt < 2; ++mt)
#pragma unroll
      for (int nt = 0; nt < 2; ++nt) {
        v8f acc = (v8f)(0.f);
#pragma unroll
        for (int kc = 0; kc < 8; ++kc) {
          int mcol = mt * 16 + nlo, k0 = kc * 4 + half * 2;
          v2f a; a.x = bufT[k0][mcol]; a.y = bufT[k0 + 1][mcol];
          acc = wmma4(a, wb0[kc][nt], acc);
        }
#pragma unroll
        for (int v = 0; v < 8; ++v) {
          int m = mt * 16 + v + half * 8, n = nt * 16 + nlo;
          if (m < 22 && n < 22) bufR[pl][m][n] = acc[v];
        }
      }
    __syncthreads();
  }

  float ss0 = 0.f, cs0 = 0.f;
  for (int idx = lane; idx < 484; idx += 32) {
    int r = idx / 22, c = idx - 22 * r;
    float mu1 = bufR[0][r][c], mu2 = bufR[1][r][c];
    float s1  = bufR[2][r][c] - mu1 * mu1;
    float s2  = bufR[3][r][c] - mu2 * mu2;
    float s12 = bufR[4][r][c] - mu1 * mu2;
    float v1 = 2.f * s12 + C2v[0], v2 = s1 + s2 + C2v[0];
    cs0 += v1 / v2;
    ss0 += (2.f * mu1 * mu2 + C1v[0]) * v1 /
           ((mu1 * mu1 + mu2 * mu2 + C1v[0]) * v2);
  }
  __syncthreads();

  // ----- Level 1: pool to 16x16, conv -> 6x6 via WMMA -----
  pool2(&bufA[0][0], 36, &pA1[0][0], 18, 16, lane);
  pool2(&bufB[0][0], 36, &pB1[0][0], 18, 16, lane);
  __syncthreads();

  for (int pl = 0; pl < 5; ++pl) {
    v8f acc = (v8f)(0.f);
#pragma unroll
    for (int kc = 0; kc < 4; ++kc) {
      int k0 = kc * 4 + half * 2;
      float a1 = pA1[nlo][k0], a2 = pA1[nlo][k0 + 1];
      float b1 = pB1[nlo][k0], b2 = pB1[nlo][k0 + 1];
      acc = wmma4(plane_val(pl, a1, a2, b1, b2), wb1[kc], acc);
    }
#pragma unroll
    for (int v = 0; v < 8; ++v) bufT[v + half * 8][nlo] = acc[v];
    __syncthreads();

    v8f acc2 = (v8f)(0.f);
#pragma unroll
    for (int kc = 0; kc < 4; ++kc) {
      int k0 = kc * 4 + half * 2;
      v2f a; a.x = bufT[k0][nlo]; a.y = bufT[k0 + 1][nlo];
      acc2 = wmma4(a, wb1[kc], acc2);
    }
#pragma unroll
    for (int v = 0; v < 8; ++v) bufR[pl][v + half * 8][nlo] = acc2[v];
    __syncthreads();
  }

  float ss1 = 0.f, cs1 = 0.f;
  for (int idx = lane; idx < 36; idx += 32) {
    int r = idx / 6, c = idx - 6 * r;
    float mu1 = bufR[0][r][c], mu2 = bufR[1][r][c];
    float s1  = bufR[2][r][c] - mu1 * mu1;
    float s2  = bufR[3][r][c] - mu2 * mu2;
    float s12 = bufR[4][r][c] - mu1 * mu2;
    float v1 = 2.f * s12 + C2v[1], v2 = s1 + s2 + C2v[1];
    cs1 += v1 / v2;
    ss1 += (2.f * mu1 * mu2 + C1v[1]) * v1 /
           ((mu1 * mu1 + mu2 * mu2 + C1v[1]) * v2);
  }
  __syncthreads();

  // ----- Levels 2-4: weighted-sum SSIM (1x1 conv output) -----
  pool2(&pA1[0][0], 18, &pA2[0][0], 10, 8, lane);
  pool2(&pB1[0][0], 18, &pB2[0][0], 10, 8, lane);
  __syncthreads();
  float ss2, cs2;
  ssim_small(&pA2[0][0], &pB2[0][0], 10, 8, g8s, lane, C1v[2], C2v[2], ss2, cs2);

  pool2(&pA2[0][0], 10, &pA3[0][0], 6, 4, lane);
  pool2(&pB2[0][0], 10, &pB3[0][0], 6, 4, lane);
  __syncthreads();
  float ss3, cs3;
  ssim_small(&pA3[0][0], &pB3[0][0], 6, 4, g4s, lane, C1v[3], C2v[3], ss3, cs3);

  pool2(&pA3[0][0], 6, &pA4[0][0], 4, 2, lane);
  pool2(&pB3[0][0], 6, &pB4[0][0], 4, 2, lane);
  __syncthreads();
  float ss4, cs4;
  ssim_small(&pA4[0][0], &pB4[0][0], 4, 2, g2s, lane, C1v[4], C2v[4], ss4, cs4);

  // ----- accumulate -----
  ss0 = wave_sum(ss0); cs0 = wave_sum(cs0);
  ss1 = wave_sum(ss1); cs1 = wave_sum(cs1);
  if (lane == 0) {
    atomicAdd(&sums[0], (double)ss0); atomicAdd(&sums[5], (double)cs0);
    atomicAdd(&sums[1], (double)ss1); atomicAdd(&sums[6], (double)cs1);
    atomicAdd(&sums[2], (double)ss2); atomicAdd(&sums[7], (double)cs2);
    atomicAdd(&sums[3], (double)ss3); atomicAdd(&sums[8], (double)cs3);
    atomicAdd(&sums[4], (double)ss4); atomicAdd(&sums[9], (double)cs4);
  }
}

// ---------------------------------------------------------------------------
// Kernel 3: combine scalars exactly like the reference:
// loss = 1 - prod_{l<4} mcs_n[l]^w[l] * (mssim_n[4]^w[4])^4
// ---------------------------------------------------------------------------
__global__ void k_final(const double* __restrict__ sums, float* __restrict__ out) {
  const float w[5] = {0.0448f, 0.2856f, 0.3001f, 0.2363f, 0.1333f};
  const double cnt[5] = {46128.0 * 484.0, 46128.0 * 36.0, 46128.0, 46128.0, 46128.0};
  float ms[5], mc[5];
#pragma unroll
  for (int l = 0; l < 5; ++l) {
    ms[l] = ((float)(sums[l]     / cnt[l]) + 1.f) * 0.5f;
    mc[l] = ((float)(sums[5 + l] / cnt[l]) + 1.f) * 0.5f;
  }
  float r = 1.f;
#pragma unroll
  for (int l = 0; l < 4; ++l) r *= powf(mc[l], w[l]);
  float p2 = powf(ms[4], w[4]);
  r *= p2 * p2 * p2 * p2;
  out[0] = 1.f - r;
}

// ---------------------------------------------------------------------------
extern "C" void kernel_launch(void* const* d_in, const int* in_sizes, int n_in,
                              void* d_out, int out_size, void* d_ws, size_t ws_size,
                              hipStream_t stream) {
  (void)in_sizes; (void)n_in; (void)out_size; (void)ws_size;
  const float* y_pred = (const float*)d_in[0];
  const float* y      = (const float*)d_in[1];
  double* sums = (double*)d_ws;
  int*    flg  = (int*)((char*)d_ws + 10 * sizeof(double));

  k_init <<<1, 32, 0, stream>>>(sums, flg);
  k_flags<<<NPATCH, 32, 0, stream>>>(y, flg);
  k_main <<<NPATCH, 32, 0, stream>>>(y, y_pred, sums, flg);
  k_final<<<1, 1, 0, stream>>>(sums, (float*)d_out);
}